// FFMtorchNN_52785148068555
// MI455X (gfx1250) — compile-verified
//
#include <hip/hip_runtime.h>

// FFM forward for MI455X (gfx1250, wave32).
// y[b] = bias + x@w + 0.25 * ( sum_{s,t,l} M[b,s,t,l]*M[b,t,s,l] - q[b] )
// M_s = X_s @ V_s  computed with V_WMMA_F32_16X16X4_F32.

typedef float v2f __attribute__((ext_vector_type(2)));
typedef float v8f __attribute__((ext_vector_type(8)));

#define NFEAT   2048
#define NFIELD  8
#define NFACT   64
#define COLS    512          // NFIELD * NFACT
#define BT      16           // batch rows per workgroup
#define MSTRIDE 516          // padded LDS row stride (floats) -> bank-conflict free

// ---------------------------------------------------------------------------
// Prologue: snorm[i] = || V[i, feat2field[i], :] ||^2   (2048 floats into d_ws)
// ---------------------------------------------------------------------------
__global__ void ffm_snorm_kernel(const float* __restrict__ V,
                                 const int* __restrict__ feat2field,
                                 float* __restrict__ snorm) {
    int i = blockIdx.x * blockDim.x + threadIdx.x;
    if (i < NFEAT) {
        int f = feat2field[i];
        const float* v = V + (size_t)i * COLS + (size_t)f * NFACT;
        float s = 0.f;
        #pragma unroll 8
        for (int l = 0; l < NFACT; ++l) s += v[l] * v[l];
        snorm[i] = s;
    }
}

// ---------------------------------------------------------------------------
// Main fused kernel: one workgroup = 16 batch rows, wave s = field s.
// ---------------------------------------------------------------------------
__global__ void __launch_bounds__(256)
ffm_forward_kernel(const float* __restrict__ x,
                   const float* __restrict__ bias,
                   const float* __restrict__ w,
                   const float* __restrict__ V,
                   const float* __restrict__ snorm,
                   float* __restrict__ out) {
    extern __shared__ float Mlds[];             // [NFIELD][BT][MSTRIDE] = 258 KB
    __shared__ float red_lin [BT][17];
    __shared__ float red_q   [BT][17];
    __shared__ float red_full[BT][17];

    const int tid   = threadIdx.x;
    const int wave  = tid >> 5;                 // field s (wave32: 8 waves)
    const int lane  = tid & 31;
    const int half  = lane >> 4;                // 0: lanes 0-15, 1: lanes 16-31
    const int l16   = lane & 15;
    const int rowbase = blockIdx.x * BT;

    // ---- linear + self(q) partial sums: thread (row, p) covers 128 features ----
    {
        const int row = tid & 15;
        const int p   = tid >> 4;               // 0..15
        const float* xr = x + (size_t)(rowbase + row) * NFEAT;
        float lin = 0.f, qq = 0.f;
        const int i0 = p * (NFEAT / 16);
        #pragma unroll 4
        for (int i = i0; i < i0 + NFEAT / 16; ++i) {
            float xv = xr[i];
            lin += xv * w[i];
            qq  += xv * xv * snorm[i];
        }
        red_lin[row][p] = lin;
        red_q  [row][p] = qq;
    }

    // ---- load A fragments once: x[rowbase+0..15, 256*s .. 256*s+255] ----
    // A (16x4 f32) layout: lane<16 -> row=lane, K={0,1}; lane>=16 -> row=lane-16, K={2,3}
    v2f afrag[64];
    {
        const float* xa = x + (size_t)(rowbase + l16) * NFEAT
                            + (size_t)wave * 256 + 2 * half;
        #pragma unroll
        for (int kt = 0; kt < 64; ++kt) {
            afrag[kt] = *(const v2f*)(xa + 4 * kt);   // 8B-aligned b64 load
        }
    }

    // ---- GEMM: M_s[16 x 512] = X_s[16 x 256] @ V_s[256 x 512] ----
    for (int nt = 0; nt < 32; ++nt) {
        const int n0 = nt * 16;
        // B (4x16 f32) layout: VGPR0 -> K row 0 (lanes<16) / 2 (lanes>=16), VGPR1 -> K row 1 / 3
        const float* vb = V + ((size_t)wave * 256 + 2 * half) * COLS + n0 + l16;
        v8f acc = {};
        #pragma unroll
        for (int kt = 0; kt < 64; ++kt) {
            const float* vk = vb + (size_t)(4 * kt) * COLS;
            v2f bfrag;
            bfrag.x = vk[0];                    // K = 4*kt + 2*half
            bfrag.y = vk[COLS];                 // K = 4*kt + 2*half + 1
            acc = __builtin_amdgcn_wmma_f32_16x16x4_f32(
                      /*neg_a=*/false, afrag[kt],
                      /*neg_b=*/false, bfrag,
                      /*c_mod=*/(short)0, acc,
                      /*reuse_a=*/false, /*reuse_b=*/false);
        }
        // C/D layout: VGPR j -> M=j (lanes<16), M=j+8 (lanes>=16); N = lane%16
        float* mbase = Mlds + ((size_t)wave * BT) * MSTRIDE + n0 + l16;
        #pragma unroll
        for (int j = 0; j < 8; ++j) {
            int mrow = j + 8 * half;
            mbase[(size_t)mrow * MSTRIDE] = acc[j];
        }
    }
    __syncthreads();

    // ---- contraction: full[b] = sum_{s,t,l} M[s][b][t*64+l] * M[t][b][s*64+l] ----
    {
        const int row = tid & 15;
        const int p   = tid >> 4;               // 4 (s,t) pairs per thread
        float fsum = 0.f;
        #pragma unroll
        for (int pi = 0; pi < 4; ++pi) {
            const int pair = p * 4 + pi;
            const int s = pair >> 3;
            const int t = pair & 7;
            const float* m_st = Mlds + ((size_t)(s * BT) + row) * MSTRIDE + t * NFACT;
            const float* m_ts = Mlds + ((size_t)(t * BT) + row) * MSTRIDE + s * NFACT;
            #pragma unroll 8
            for (int l = 0; l < NFACT; ++l) fsum += m_st[l] * m_ts[l];
        }
        red_full[row][p] = fsum;
    }
    __syncthreads();

    // ---- final reduce + epilogue ----
    if (tid < BT) {
        const int row = tid;
        float lin = 0.f, qq = 0.f, fl = 0.f;
        #pragma unroll
        for (int p = 0; p < 16; ++p) {
            lin += red_lin[row][p];
            qq  += red_q  [row][p];
            fl  += red_full[row][p];
        }
        out[rowbase + row] = bias[0] + lin + 0.25f * (fl - qq);
    }
}

// ---------------------------------------------------------------------------
extern "C" void kernel_launch(void* const* d_in, const int* in_sizes, int n_in,
                              void* d_out, int out_size, void* d_ws, size_t ws_size,
                              hipStream_t stream) {
    const float* x    = (const float*)d_in[0];   // [8192, 2048]
    const float* b    = (const float*)d_in[1];   // [1]
    const float* w    = (const float*)d_in[2];   // [2048, 1]
    const float* V    = (const float*)d_in[3];   // [2048, 8, 64]
    const int*   f2f  = (const int*)  d_in[4];   // [2048]
    float*       out  = (float*)d_out;           // [8192]
    float*       snorm = (float*)d_ws;           // 2048 floats of scratch

    hipLaunchKernelGGL(ffm_snorm_kernel, dim3((NFEAT + 255) / 256), dim3(256), 0, stream,
                       V, f2f, snorm);

    const int nblocks = out_size / BT;           // 8192 / 16 = 512
    const size_t shmem = (size_t)NFIELD * BT * MSTRIDE * sizeof(float); // 258 KB
    hipLaunchKernelGGL(ffm_forward_kernel, dim3(nblocks), dim3(256), shmem, stream,
                       x, b, w, V, snorm, out);
}